// LengthRegulator_5669356833183
// MI455X (gfx1250) — compile-verified
//
#include <hip/hip_runtime.h>

// Problem constants (from reference)
#define BATCH 32
#define TMAXC 512
#define DFEAT 384            // floats per row (1536 bytes)
#define TOUTC 3584           // TMAX * (MAXDUR-1)
#define ROWS  (BATCH * TOUTC) // 114688 output rows

#define WPB 8                // waves per block (256 threads, wave32)

typedef __attribute__((ext_vector_type(4))) float f32x4;
typedef __attribute__((ext_vector_type(4))) int   i32x4;

typedef __attribute__((address_space(1))) i32x4 gi32x4;  // global
typedef __attribute__((address_space(3))) i32x4 li32x4;  // LDS

// ---------------------------------------------------------------------------
// Kernel 1: per-batch masked cumsum + scatter frame->token index map.
// One block (512 threads) per batch. idxmap[b*TOUT+f] = token index, or -1=pad.
// ---------------------------------------------------------------------------
__global__ __launch_bounds__(TMAXC) void lr_build_idx(
    const int* __restrict__ ds, const int* __restrict__ ilens,
    int* __restrict__ idxmap)
{
    __shared__ int cum[TMAXC];
    const int b = blockIdx.x;
    const int t = threadIdx.x;

    const int ilen = ilens[b];
    int d = (t < ilen) ? ds[b * TMAXC + t] : 0;

    // Block-wide inclusive scan (Hillis-Steele) over 512 elements in LDS.
    cum[t] = d;
    __syncthreads();
    for (int off = 1; off < TMAXC; off <<= 1) {
        int v   = cum[t];
        int add = (t >= off) ? cum[t - off] : 0;
        __syncthreads();
        cum[t] = v + add;
        __syncthreads();
    }
    const int total = cum[TMAXC - 1];
    __syncthreads();

    int start;
    if (total == 0) {
        // all-zero fallback: d = mask -> every valid token has duration 1,
        // cum[t] = t+1 for t<ilen, so start == t analytically.
        d = (t < ilen) ? 1 : 0;
        start = t;
    } else {
        start = cum[t] - d;
    }

    // Initialize whole batch row map to pad (-1), then scatter token indices.
    for (int f = t; f < TOUTC; f += TMAXC)
        idxmap[b * TOUTC + f] = -1;
    __syncthreads();   // order init writes before scatter writes (same block)

    for (int k = 0; k < d; ++k)
        idxmap[b * TOUTC + start + k] = t;   // start+d <= total <= TOUT
}

// ---------------------------------------------------------------------------
// Kernel 2: bandwidth streamer. One wave per output row, double-buffered
// async global->LDS loads (CDNA5 async-tensor path), non-temporal B128 stores.
// ---------------------------------------------------------------------------
__global__ __launch_bounds__(WPB * 32) void lr_gather(
    const float* __restrict__ xs, const int* __restrict__ idxmap,
    float* __restrict__ out, int totalWaves)
{
    // 8 waves * 2 buffers * 384 floats = 24 KB LDS per block.
    __shared__ __align__(16) float lds[WPB * 2 * DFEAT];

    const int wave = threadIdx.x >> 5;
    const int lane = threadIdx.x & 31;
    const int gw   = blockIdx.x * WPB + wave;

    float* wbuf = &lds[wave * (2 * DFEAT)];

    // Issue one row's 3 async b128 chunks (512 B each) into buffer `buf`.
    auto issue_row = [&](int r, int idx, int buf) {
        const int b = r / TOUTC;
        const float* src = xs + ((size_t)b * TMAXC + (size_t)idx) * DFEAT + lane * 4;
        gi32x4* gp = (gi32x4*)src;                                 // drops const, adds AS1
        li32x4* lp = (li32x4*)(wbuf + buf * DFEAT + lane * 4);     // AS3
        // INST_OFFSET is added to both the global and the LDS address.
        __builtin_amdgcn_global_load_async_to_lds_b128(gp, lp, 0,    0);
        __builtin_amdgcn_global_load_async_to_lds_b128(gp, lp, 512,  0);
        __builtin_amdgcn_global_load_async_to_lds_b128(gp, lp, 1024, 0);
    };

    int r   = gw;
    int buf = 0;
    int idx = -1;
    if (r < ROWS) {
        idx = __builtin_amdgcn_readfirstlane(idxmap[r]);  // wave-uniform
        if (idx >= 0) issue_row(r, idx, buf);
    }

    while (r < ROWS) {
        const int rn = r + totalWaves;
        int idxn = -1;
        if (rn < ROWS)
            idxn = __builtin_amdgcn_readfirstlane(idxmap[rn]);
        if (idxn >= 0) issue_row(rn, idxn, buf ^ 1);      // prefetch next row

        f32x4 v0, v1, v2;
        if (idx >= 0) {
            // Wait only for *this* row's 3 loads; leave the prefetch in flight.
            if (idxn >= 0) __builtin_amdgcn_s_wait_asynccnt(3);
            else           __builtin_amdgcn_s_wait_asynccnt(0);
            const f32x4* lsrc = (const f32x4*)(wbuf + buf * DFEAT);
            v0 = lsrc[lane];
            v1 = lsrc[lane + 32];
            v2 = lsrc[lane + 64];
        } else {
            v0 = f32x4{0.f, 0.f, 0.f, 0.f};
            v1 = v0;
            v2 = v0;
        }

        float* dst = out + (size_t)r * DFEAT + lane * 4;
        // Output is write-once / never re-read: non-temporal to spare L2.
        __builtin_nontemporal_store(v0, (f32x4*)dst);
        __builtin_nontemporal_store(v1, (f32x4*)(dst + 128));
        __builtin_nontemporal_store(v2, (f32x4*)(dst + 256));

        r = rn; idx = idxn; buf ^= 1;
    }
}

// ---------------------------------------------------------------------------
// Launcher. Inputs (setup_inputs order): xs f32[32,512,384], ds i32[32,512],
// ilens i32[32]. Output f32[32,3584,384]. Workspace: idxmap needs
// 32*3584*4 = 458752 bytes.
// ---------------------------------------------------------------------------
extern "C" void kernel_launch(void* const* d_in, const int* in_sizes, int n_in,
                              void* d_out, int out_size, void* d_ws, size_t ws_size,
                              hipStream_t stream) {
    const float* xs     = (const float*)d_in[0];
    const int*   ds     = (const int*)d_in[1];
    const int*   ilens  = (const int*)d_in[2];
    float*       out    = (float*)d_out;
    int*         idxmap = (int*)d_ws;   // 458752 B required

    lr_build_idx<<<BATCH, TMAXC, 0, stream>>>(ds, ilens, idxmap);

    const int blocks = 1792;            // 14336 waves -> 8 rows per wave
    lr_gather<<<blocks, WPB * 32, 0, stream>>>(xs, idxmap, out, blocks * WPB);
}